// GATNet_61950608278030
// MI455X (gfx1250) — compile-verified
//
#include <hip/hip_runtime.h>
#include <hip/hip_bf16.h>

#define NN 50000
#define NE 1600000

typedef __attribute__((ext_vector_type(2))) float v2f;
typedef __attribute__((ext_vector_type(8))) float v8f;

// order-preserving float <-> uint mapping for atomic max (deterministic max)
__device__ __forceinline__ unsigned fenc(float f) {
    unsigned u = __float_as_uint(f);
    return (u & 0x80000000u) ? ~u : (u | 0x80000000u);
}
__device__ __forceinline__ float fdec(unsigned k) {
    return __uint_as_float((k & 0x80000000u) ? (k & 0x7FFFFFFFu) : ~k);
}
__device__ __forceinline__ float leaky(float v) { return v > 0.f ? v : 0.2f * v; }

// ---------------------------------------------------------------------------
// GEMM1: h1[50000,64] = x[50000,512] @ W1[512,64]   (fp32 WMMA 16x16x4)
// block = 128 threads = 4 waves; wave w owns N-tile [16w,16w+16); 16 rows/block
// ---------------------------------------------------------------------------
__global__ __launch_bounds__(128) void gemm1_wmma(const float* __restrict__ x,
                                                  const float* __restrict__ W1,
                                                  float* __restrict__ h1) {
    __shared__ float As[16 * 132];   // 16 x 128 (+4 pad)
    __shared__ float Ws[128 * 68];   // 128 x 64 (+4 pad)
    const int row0 = blockIdx.x * 16;
    const int tid  = threadIdx.x;
    const int wv   = tid >> 5;
    const int lane = tid & 31;
    const int lhi  = lane >> 4;      // 0/1 half-wave
    const int llo  = lane & 15;
    v8f acc = {};
    for (int kb = 0; kb < 512; kb += 128) {
        for (int q = tid; q < 512; q += 128) {       // A tile: 512 float4 slots
            int f = q * 4, m = f >> 7, k = f & 127;
            *(float4*)(&As[m * 132 + k]) =
                *(const float4*)(x + (size_t)(row0 + m) * 512 + kb + k);
        }
        for (int q = tid; q < 2048; q += 128) {      // W tile: 2048 float4 slots
            int f = q * 4, k = f >> 6, n = f & 63;
            *(float4*)(&Ws[k * 68 + n]) =
                *(const float4*)(W1 + (size_t)(kb + k) * 64 + n);
        }
        __syncthreads();
#pragma unroll 4
        for (int ks = 0; ks < 128; ks += 4) {
            const int kk = ks + (lhi << 1);
            v2f a = *(const v2f*)(&As[llo * 132 + kk]);      // A[m=llo][kk,kk+1]
            v2f b;
            b.x = Ws[kk * 68 + 16 * wv + llo];               // B[kk][n]
            b.y = Ws[(kk + 1) * 68 + 16 * wv + llo];         // B[kk+1][n]
            acc = __builtin_amdgcn_wmma_f32_16x16x4_f32(false, a, false, b,
                                                        (short)0, acc, false, false);
        }
        __syncthreads();
    }
    const int n = 16 * wv + llo;
#pragma unroll
    for (int v = 0; v < 8; ++v) {
        int m = v + 8 * lhi;
        h1[(size_t)(row0 + m) * 64 + n] = acc[v];
    }
}

// ---------------------------------------------------------------------------
// GEMM2: h2[50000,40] = hp[50000,64] @ W2[64,40]  (N padded to 48, 3 waves)
// ---------------------------------------------------------------------------
__global__ __launch_bounds__(96) void gemm2_wmma(const float* __restrict__ hp,
                                                 const float* __restrict__ W2,
                                                 float* __restrict__ h2) {
    __shared__ float As[16 * 68];    // 16 x 64 (+4 pad)
    __shared__ float Ws[64 * 52];    // 64 x 48 (+4 pad)
    const int row0 = blockIdx.x * 16;
    const int tid  = threadIdx.x;
    const int wv   = tid >> 5;
    const int lane = tid & 31;
    const int lhi  = lane >> 4;
    const int llo  = lane & 15;
    for (int q = tid; q < 256; q += 96) {            // A: 256 float4 slots
        int f = q * 4, m = f >> 6, k = f & 63;
        *(float4*)(&As[m * 68 + k]) =
            *(const float4*)(hp + (size_t)(row0 + m) * 64 + k);
    }
    for (int q = tid; q < 64 * 48; q += 96) {        // W2 padded with zeros
        int k = q / 48, n = q % 48;
        Ws[k * 52 + n] = (n < 40) ? W2[k * 40 + n] : 0.f;
    }
    __syncthreads();
    v8f acc = {};
#pragma unroll
    for (int ks = 0; ks < 64; ks += 4) {
        const int kk = ks + (lhi << 1);
        v2f a = *(const v2f*)(&As[llo * 68 + kk]);
        v2f b;
        b.x = Ws[kk * 52 + 16 * wv + llo];
        b.y = Ws[(kk + 1) * 52 + 16 * wv + llo];
        acc = __builtin_amdgcn_wmma_f32_16x16x4_f32(false, a, false, b,
                                                    (short)0, acc, false, false);
    }
    const int n = 16 * wv + llo;
    if (n < 40) {
#pragma unroll
        for (int v = 0; v < 8; ++v) {
            int m = v + 8 * lhi;
            h2[(size_t)(row0 + m) * 40 + n] = acc[v];
        }
    }
}

// --------------------------- attention scores ------------------------------
__global__ void score1_kernel(const float* __restrict__ h1,
                              const float* __restrict__ a_src,
                              const float* __restrict__ a_dst,
                              float* __restrict__ es, float* __restrict__ ed) {
    int idx = blockIdx.x * blockDim.x + threadIdx.x;
    if (idx >= NN * 8) return;
    int n = idx >> 3, h = idx & 7;
    const float4* hv = (const float4*)(h1 + (size_t)n * 64 + h * 8);
    const float4* as = (const float4*)(a_src + h * 8);
    const float4* ad = (const float4*)(a_dst + h * 8);
    float4 h0 = hv[0], h1v = hv[1];
    float4 s0 = as[0], s1 = as[1], d0 = ad[0], d1 = ad[1];
    es[idx] = h0.x*s0.x + h0.y*s0.y + h0.z*s0.z + h0.w*s0.w
            + h1v.x*s1.x + h1v.y*s1.y + h1v.z*s1.z + h1v.w*s1.w;
    ed[idx] = h0.x*d0.x + h0.y*d0.y + h0.z*d0.z + h0.w*d0.w
            + h1v.x*d1.x + h1v.y*d1.y + h1v.z*d1.z + h1v.w*d1.w;
}

__global__ void score2_kernel(const float* __restrict__ h2,
                              const float* __restrict__ a_src,
                              const float* __restrict__ a_dst,
                              float* __restrict__ es, float* __restrict__ ed) {
    int n = blockIdx.x * blockDim.x + threadIdx.x;
    if (n >= NN) return;
    float s = 0.f, d = 0.f;
    for (int c = 0; c < 40; ++c) {
        float v = h2[(size_t)n * 40 + c];
        s += v * a_src[c];
        d += v * a_dst[c];
    }
    es[n] = s; ed[n] = d;
}

// --------------------------- layer 1 edge passes ---------------------------
__global__ void edge1_max(const int* __restrict__ src, const int* __restrict__ dst,
                          const float* __restrict__ es, const float* __restrict__ ed,
                          unsigned* __restrict__ keys) {
    int e = blockIdx.x * blockDim.x + threadIdx.x;
    if (e >= NE) return;
    int s = src[e], d = dst[e];
    const float4* ps = (const float4*)(es + (size_t)s * 8);
    const float4* pd = (const float4*)(ed + (size_t)d * 8);
    float4 a0 = ps[0], a1 = ps[1], b0 = pd[0], b1 = pd[1];
    float v[8] = {a0.x + b0.x, a0.y + b0.y, a0.z + b0.z, a0.w + b0.w,
                  a1.x + b1.x, a1.y + b1.y, a1.z + b1.z, a1.w + b1.w};
    unsigned* kb = keys + (size_t)d * 8;
#pragma unroll
    for (int h = 0; h < 8; ++h) atomicMax(kb + h, fenc(leaky(v[h])));
}

__global__ void edge1_sum(const int* __restrict__ src, const int* __restrict__ dst,
                          const float* __restrict__ es, const float* __restrict__ ed,
                          const unsigned* __restrict__ keys, float* __restrict__ denom) {
    int e = blockIdx.x * blockDim.x + threadIdx.x;
    if (e >= NE) return;
    int s = src[e], d = dst[e];
    const float4* ps = (const float4*)(es + (size_t)s * 8);
    const float4* pd = (const float4*)(ed + (size_t)d * 8);
    float4 a0 = ps[0], a1 = ps[1], b0 = pd[0], b1 = pd[1];
    float v[8] = {a0.x + b0.x, a0.y + b0.y, a0.z + b0.z, a0.w + b0.w,
                  a1.x + b1.x, a1.y + b1.y, a1.z + b1.z, a1.w + b1.w};
#pragma unroll
    for (int h = 0; h < 8; ++h) {
        float m = fdec(keys[(size_t)d * 8 + h]);
        atomicAdd(denom + (size_t)d * 8 + h, expf(leaky(v[h]) - m));
    }
}

__global__ void edge1_agg(const int* __restrict__ src, const int* __restrict__ dst,
                          const float* __restrict__ es, const float* __restrict__ ed,
                          const unsigned* __restrict__ keys, const float* __restrict__ denom,
                          const float* __restrict__ h1, float* __restrict__ agg) {
    int e = blockIdx.x * blockDim.x + threadIdx.x;
    if (e >= NE) return;
    int s = src[e], d = dst[e];
    const float4* ps = (const float4*)(es + (size_t)s * 8);
    const float4* pd = (const float4*)(ed + (size_t)d * 8);
    float4 a0 = ps[0], a1 = ps[1], b0 = pd[0], b1 = pd[1];
    float v[8] = {a0.x + b0.x, a0.y + b0.y, a0.z + b0.z, a0.w + b0.w,
                  a1.x + b1.x, a1.y + b1.y, a1.z + b1.z, a1.w + b1.w};
    float al[8];
#pragma unroll
    for (int h = 0; h < 8; ++h) {
        float m  = fdec(keys[(size_t)d * 8 + h]);
        float ex = expf(leaky(v[h]) - m);
        al[h] = ex / denom[(size_t)d * 8 + h];
    }
    const float4* hs = (const float4*)(h1 + (size_t)s * 64);
    float* ag = agg + (size_t)d * 64;
#pragma unroll
    for (int q = 0; q < 16; ++q) {       // feature group f = 4q..4q+3, head = q>>1
        float4 hv = hs[q];
        float a = al[q >> 1];
        atomicAdd(ag + q * 4 + 0, hv.x * a);
        atomicAdd(ag + q * 4 + 1, hv.y * a);
        atomicAdd(ag + q * 4 + 2, hv.z * a);
        atomicAdd(ag + q * 4 + 3, hv.w * a);
    }
}

// --------------------------- bias + ELU ------------------------------------
__global__ void elu_kernel(const float* __restrict__ agg, const float* __restrict__ b1,
                           float* __restrict__ hp) {
    int i = blockIdx.x * blockDim.x + threadIdx.x;
    if (i >= NN * 64) return;
    float v = agg[i] + b1[i & 63];
    hp[i] = v > 0.f ? v : expm1f(v);
}

// --------------------------- layer 2 edge passes (H=1, C=40) ---------------
__global__ void edge2_max(const int* __restrict__ src, const int* __restrict__ dst,
                          const float* __restrict__ es, const float* __restrict__ ed,
                          unsigned* __restrict__ keys) {
    int e = blockIdx.x * blockDim.x + threadIdx.x;
    if (e >= NE) return;
    int s = src[e], d = dst[e];
    atomicMax(keys + d, fenc(leaky(es[s] + ed[d])));
}

__global__ void edge2_sum(const int* __restrict__ src, const int* __restrict__ dst,
                          const float* __restrict__ es, const float* __restrict__ ed,
                          const unsigned* __restrict__ keys, float* __restrict__ denom) {
    int e = blockIdx.x * blockDim.x + threadIdx.x;
    if (e >= NE) return;
    int s = src[e], d = dst[e];
    float m = fdec(keys[d]);
    atomicAdd(denom + d, expf(leaky(es[s] + ed[d]) - m));
}

__global__ void edge2_agg(const int* __restrict__ src, const int* __restrict__ dst,
                          const float* __restrict__ es, const float* __restrict__ ed,
                          const unsigned* __restrict__ keys, const float* __restrict__ denom,
                          const float* __restrict__ h2, float* __restrict__ agg) {
    int e = blockIdx.x * blockDim.x + threadIdx.x;
    if (e >= NE) return;
    int s = src[e], d = dst[e];
    float m  = fdec(keys[d]);
    float al = expf(leaky(es[s] + ed[d]) - m) / denom[d];
    const float4* hs = (const float4*)(h2 + (size_t)s * 40);
    float* ag = agg + (size_t)d * 40;
#pragma unroll
    for (int q = 0; q < 10; ++q) {
        float4 hv = hs[q];
        atomicAdd(ag + q * 4 + 0, hv.x * al);
        atomicAdd(ag + q * 4 + 1, hv.y * al);
        atomicAdd(ag + q * 4 + 2, hv.z * al);
        atomicAdd(ag + q * 4 + 3, hv.w * al);
    }
}

// --------------------------- bias + log_softmax ----------------------------
__global__ void finalize_kernel(const float* __restrict__ agg, const float* __restrict__ b2,
                                float* __restrict__ out) {
    int n = blockIdx.x * blockDim.x + threadIdx.x;
    if (n >= NN) return;
    float buf[40];
    float mx = -3.402823466e38f;
    for (int c = 0; c < 40; ++c) {
        float v = agg[(size_t)n * 40 + c] + b2[c];
        buf[c] = v;
        mx = fmaxf(mx, v);
    }
    float sum = 0.f;
    for (int c = 0; c < 40; ++c) sum += expf(buf[c] - mx);
    float l = logf(sum);
    for (int c = 0; c < 40; ++c) out[(size_t)n * 40 + c] = buf[c] - mx - l;
}

// ---------------------------------------------------------------------------
extern "C" void kernel_launch(void* const* d_in, const int* in_sizes, int n_in,
                              void* d_out, int out_size, void* d_ws, size_t ws_size,
                              hipStream_t stream) {
    const float* x      = (const float*)d_in[0];
    const float* W1     = (const float*)d_in[1];
    const float* a_src1 = (const float*)d_in[2];
    const float* a_dst1 = (const float*)d_in[3];
    const float* b1     = (const float*)d_in[4];
    const float* W2     = (const float*)d_in[5];
    const float* a_src2 = (const float*)d_in[6];
    const float* a_dst2 = (const float*)d_in[7];
    const float* b2     = (const float*)d_in[8];
    const int*   esrc   = (const int*)d_in[9];
    const int*   edst   = (const int*)d_in[10];
    float*       out    = (float*)d_out;

    // workspace layout (floats); total 15.4M floats = 61.6 MB
    float* W = (float*)d_ws;
    float*    h1     = W;                  // 3,200,000
    float*    hp     = W + 3200000;        // 3,200,000
    float*    h2     = W + 6400000;        // 2,000,000
    float*    es1    = W + 8400000;        //   400,000
    float*    ed1    = W + 8800000;        //   400,000
    float*    es2    = W + 9200000;        //    50,000
    float*    ed2    = W + 9250000;        //    50,000
    float*    zbase  = W + 9300000;        // zeroed region start
    unsigned* keys1  = (unsigned*)(W + 9300000);   //   400,000
    float*    denom1 = W + 9700000;                //   400,000
    float*    agg1   = W + 10100000;               // 3,200,000
    unsigned* keys2  = (unsigned*)(W + 13300000);  //    50,000
    float*    denom2 = W + 13350000;               //    50,000
    float*    agg2   = W + 13400000;               // 2,000,000 -> 15,400,000

    hipMemsetAsync(zbase, 0, (size_t)(15400000 - 9300000) * sizeof(float), stream);

    gemm1_wmma<<<NN / 16, 128, 0, stream>>>(x, W1, h1);
    score1_kernel<<<(NN * 8 + 255) / 256, 256, 0, stream>>>(h1, a_src1, a_dst1, es1, ed1);
    edge1_max<<<NE / 256, 256, 0, stream>>>(esrc, edst, es1, ed1, keys1);
    edge1_sum<<<NE / 256, 256, 0, stream>>>(esrc, edst, es1, ed1, keys1, denom1);
    edge1_agg<<<NE / 256, 256, 0, stream>>>(esrc, edst, es1, ed1, keys1, denom1, h1, agg1);
    elu_kernel<<<(NN * 64) / 256, 256, 0, stream>>>(agg1, b1, hp);
    gemm2_wmma<<<NN / 16, 96, 0, stream>>>(hp, W2, h2);
    score2_kernel<<<(NN + 255) / 256, 256, 0, stream>>>(h2, a_src2, a_dst2, es2, ed2);
    edge2_max<<<NE / 256, 256, 0, stream>>>(esrc, edst, es2, ed2, keys2);
    edge2_sum<<<NE / 256, 256, 0, stream>>>(esrc, edst, es2, ed2, keys2, denom2);
    edge2_agg<<<NE / 256, 256, 0, stream>>>(esrc, edst, es2, ed2, keys2, denom2, h2, agg2);
    finalize_kernel<<<(NN + 255) / 256, 256, 0, stream>>>(agg2, b2, out);
}